// multihead_attention_87454124081549
// MI455X (gfx1250) — compile-verified
//
#include <hip/hip_runtime.h>
#include <hip/hip_bf16.h>

// CDNA5 / gfx1250, wave32. All GEMMs go through V_WMMA_F32_16X16X4_F32.
typedef float v2f __attribute__((ext_vector_type(2)));
typedef float v8f __attribute__((ext_vector_type(8)));

#define WMMA_F32(a, b, c) \
  __builtin_amdgcn_wmma_f32_16x16x4_f32(false, (a), false, (b), (short)0, (c), false, false)

static constexpr int   NU   = 256;
static constexpr int   NH   = 32;
static constexpr int   B    = 128;
static constexpr int   S    = 256;
static constexpr int   NHEAD= NU / NH;             // 8
static constexpr int   NN   = NHEAD * B;           // 1024 head-batch rows
static constexpr size_t PROJ = (size_t)B * S * NU; // 8388608 floats per tensor

// Branchless tanh via the HW transcendental (v_exp_f32). Clamp avoids inf/inf;
// tanh(15) == 1.0f in fp32 anyway. No EXEC-mask churn, ~6 VALU ops.
__device__ __forceinline__ float fast_tanh(float x) {
  const float xc = fminf(fmaxf(x, -15.0f), 15.0f);
  const float e  = __expf(2.0f * xc);
  return (e - 1.0f) / (e + 1.0f);
}

// ---------------------------------------------------------------------------
// Kernel 1: Y = tanh(X @ W^T + b), output written in head-split layout
//   out[((col/32)*B + b)*S*NH + s*NH + (col%32)],  input row r = b*S + s.
// Optional adj: row r reads X[adj[b]*S + s] (folds the queries4 gather in).
// Each wave computes a 16-row x 64-col strip (4 accumulators reuse the A frag).
// ---------------------------------------------------------------------------
__global__ void proj_tanh_headsplit(const float* __restrict__ X,
                                    const int*   __restrict__ adj,
                                    const float* __restrict__ W,
                                    const float* __restrict__ bias,
                                    float* __restrict__ out) {
  const int lane = threadIdx.x & 31;
  const int wave = threadIdx.x >> 5;
  const int wt   = blockIdx.x * 8 + wave;   // 8192 wave-tiles (2048 row-tiles x 4 col-groups)
  const int tm   = wt >> 2;                 // row tile (16 rows)
  const int tg   = wt & 3;                  // 64-col group
  const int half = lane >> 4;
  const int l16  = lane & 15;

  const int row  = tm * 16 + l16;           // A-matrix row for this lane
  const int bb   = row >> 8;                // batch
  const int ss   = row & 255;               // seq
  const int srcb = adj ? adj[bb] : bb;
  const float* xrow = X + ((size_t)srcb * S + ss) * NU;
  const float* w0   = W + (size_t)(tg * 64 + l16) * NU;  // W row (output col) for B frag

  v8f acc[4] = {};
  for (int k0 = 0; k0 < NU; k0 += 4) {
    v2f a = *(const v2f*)(xrow + k0 + 2 * half);          // A: lanes 0-15 K0/K1, 16-31 K2/K3
#pragma unroll
    for (int t = 0; t < 4; ++t) {
      v2f b = *(const v2f*)(w0 + (size_t)t * 16 * NU + k0 + 2 * half);
      acc[t] = WMMA_F32(a, b, acc[t]);
    }
  }
#pragma unroll
  for (int t = 0; t < 4; ++t) {
    const int col = tg * 64 + t * 16 + l16;
    const float bv = bias[col];
    const int c = col >> 5, h = col & 31;
#pragma unroll
    for (int i = 0; i < 8; ++i) {
      const int m  = tm * 16 + i + 8 * half;              // D row: lanes<16 -> i, else i+8
      const int ob = m >> 8, os = m & 255;
      out[(((size_t)c * B + ob) * S + os) * NH + h] = fast_tanh(acc[t][i] + bv);
    }
  }
}

// ---------------------------------------------------------------------------
// Kernel 2: softmax stats over axis n (online over 1024 head-batch rows).
// For each branch br and (q,k) 16x16 tile: M[q,k]=max_n s, Zr=1/sum exp(s-M),
// s = Q[n,q,:].K[n,k,:]/sqrt(32). Scores never hit memory; reciprocal stored
// so the output pass is multiply-only.
// ---------------------------------------------------------------------------
__global__ void attn_stats(const float* __restrict__ ws_qkv,
                           float* __restrict__ Mout,
                           float* __restrict__ Zrout) {
  const int lane = threadIdx.x & 31;
  const int wave = threadIdx.x >> 5;
  const int wt   = blockIdx.x * 8 + wave;   // 1024 wave-tiles: 4 br x 16 qt x 16 kt
  const int br   = wt >> 8;
  const int qt   = (wt >> 4) & 15;
  const int kt   = wt & 15;
  const int half = lane >> 4, l16 = lane & 15;

  const float* Q = ws_qkv + (size_t)br * PROJ;
  const float* K = ws_qkv + (size_t)4 * PROJ;
  const float scale = 0.17677669529663687f;  // 1/sqrt(32)

  v8f m, z;
#pragma unroll
  for (int i = 0; i < 8; ++i) { m[i] = -3.0e38f; z[i] = 0.0f; }

  for (int n = 0; n < NN; ++n) {
    const float* Qn = Q + (size_t)n * (S * NH) + (size_t)(qt * 16 + l16) * NH;
    const float* Kn = K + (size_t)n * (S * NH) + (size_t)(kt * 16 + l16) * NH;
    v8f s = {};
#pragma unroll
    for (int d0 = 0; d0 < NH; d0 += 4) {
      v2f a = *(const v2f*)(Qn + d0 + 2 * half);
      v2f b = *(const v2f*)(Kn + d0 + 2 * half);
      s = WMMA_F32(a, b, s);
    }
#pragma unroll
    for (int i = 0; i < 8; ++i) {
      const float sv = s[i] * scale;
      const float mn = fmaxf(m[i], sv);
      z[i] = z[i] * __expf(m[i] - mn) + __expf(sv - mn);
      m[i] = mn;
    }
  }
#pragma unroll
  for (int i = 0; i < 8; ++i) {
    const int q = qt * 16 + i + 8 * half;
    const int k = kt * 16 + l16;
    const size_t idx = (size_t)br * (S * S) + (size_t)q * S + k;
    Mout[idx]  = m[i];
    Zrout[idx] = 1.0f / z[i];
  }
}

// ---------------------------------------------------------------------------
// Kernel 3: attention output. Recompute s = QK^T per (br,n,q-tile), apply
// p = exp(s*scale - M)*Zr, stage p through LDS (C/D layout -> A layout), then
// o += p @ V via WMMA. Writes merged-head layout directly into d_out slots
// (o1,o2,o3) and workspace (o4).
// ---------------------------------------------------------------------------
__global__ void attn_out(const float* __restrict__ ws_qkv,
                         const float* __restrict__ Mbuf,
                         const float* __restrict__ Zrbuf,
                         float* __restrict__ d_out,
                         float* __restrict__ o4ws) {
  __shared__ float plds[8][16 * 17];        // padded 16x16 p tile per wave
  const int lane = threadIdx.x & 31;
  const int wave = threadIdx.x >> 5;
  const int wt   = blockIdx.x * 8 + wave;   // 65536 wave-tiles: 4 br x 1024 n x 16 qt
  const int br   = wt >> 14;
  const int n    = (wt >> 4) & 1023;
  const int qt   = wt & 15;
  const int half = lane >> 4, l16 = lane & 15;

  const float* Q  = ws_qkv + (size_t)br * PROJ + (size_t)n * (S * NH);
  const float* K  = ws_qkv + (size_t)4 * PROJ + (size_t)n * (S * NH);
  const float* V  = ws_qkv + (size_t)(5 + br) * PROJ + (size_t)n * (S * NH);
  const float* Mp = Mbuf  + (size_t)br * (S * S);
  const float* Zp = Zrbuf + (size_t)br * (S * S);
  const float scale = 0.17677669529663687f;
  float* p = &plds[wave][0];

  v8f o_lo = {}, o_hi = {};
  const float* Qr = Q + (size_t)(qt * 16 + l16) * NH;

  for (int kt = 0; kt < 16; ++kt) {
    // s-tile = Q (16x32) x K^T (32x16)
    v8f s = {};
    const float* Kr = K + (size_t)(kt * 16 + l16) * NH;
#pragma unroll
    for (int d0 = 0; d0 < NH; d0 += 4) {
      v2f a = *(const v2f*)(Qr + d0 + 2 * half);
      v2f b = *(const v2f*)(Kr + d0 + 2 * half);
      s = WMMA_F32(a, b, s);
    }
    // p = exp(s*scale - M) * (1/Z)  -> LDS (row-major q x k within tile)
#pragma unroll
    for (int i = 0; i < 8; ++i) {
      const int q = qt * 16 + i + 8 * half;
      const int k = kt * 16 + l16;
      const float pv = __expf(s[i] * scale - Mp[(size_t)q * S + k]) * Zp[(size_t)q * S + k];
      p[(i + 8 * half) * 17 + l16] = pv;
    }
    // o (16x32) += p (16x16) @ V-tile (16x32); same-wave DS ops are in order.
    const int vb = kt * 16;
#pragma unroll
    for (int kk = 0; kk < 16; kk += 4) {
      const int krow = kk + 2 * half;
      v2f a; a.x = p[l16 * 17 + krow]; a.y = p[l16 * 17 + krow + 1];
      v2f b0; b0.x = V[(size_t)(vb + krow) * NH + l16];
              b0.y = V[(size_t)(vb + krow + 1) * NH + l16];
      o_lo = WMMA_F32(a, b0, o_lo);
      v2f b1; b1.x = V[(size_t)(vb + krow) * NH + 16 + l16];
              b1.y = V[(size_t)(vb + krow + 1) * NH + 16 + l16];
      o_hi = WMMA_F32(a, b1, o_hi);
    }
  }
  // merge heads: n = c*B + bb ; dest row = bb*S + q, cols c*32 + [0,32)
  const int c = n >> 7, bb = n & 127;
  float* dst = (br == 0) ? d_out
             : (br == 1) ? d_out + PROJ
             : (br == 2) ? d_out + 3 * PROJ   // o3 lives in output slot 3 (o1,o2,o5,o3)
                         : o4ws;
#pragma unroll
  for (int i = 0; i < 8; ++i) {
    const int q = qt * 16 + i + 8 * half;
    const size_t base = (((size_t)bb * S + q) * NU) + (size_t)c * NH;
    dst[base + l16]      = o_lo[i];
    dst[base + 16 + l16] = o_hi[i];
  }
}

// ---------------------------------------------------------------------------
// Kernel 4: o5 = tanh(concat(o1,o2,o3,o4) @ W5^T + b5), K = 1024.
// ---------------------------------------------------------------------------
__global__ void out_proj_tanh(const float* __restrict__ o1,
                              const float* __restrict__ o2,
                              const float* __restrict__ o3,
                              const float* __restrict__ o4,
                              const float* __restrict__ W5,
                              const float* __restrict__ b5,
                              float* __restrict__ out) {
  const int lane = threadIdx.x & 31;
  const int wave = threadIdx.x >> 5;
  const int wt   = blockIdx.x * 8 + wave;   // 8192 wave-tiles (2048 row-tiles x 4 col-groups)
  const int tm   = wt >> 2;
  const int tg   = wt & 3;
  const int half = lane >> 4, l16 = lane & 15;

  const float* srcs[4] = {o1, o2, o3, o4};
  const size_t arow = (size_t)(tm * 16 + l16) * NU;
  const float* w0   = W5 + (size_t)(tg * 64 + l16) * (4 * NU);

  v8f acc[4] = {};
#pragma unroll
  for (int part = 0; part < 4; ++part) {
    const float* xrow  = srcs[part] + arow;
    const float* wpart = w0 + part * NU;
    for (int k0 = 0; k0 < NU; k0 += 4) {
      v2f a = *(const v2f*)(xrow + k0 + 2 * half);
#pragma unroll
      for (int t = 0; t < 4; ++t) {
        v2f b = *(const v2f*)(wpart + (size_t)t * 16 * (4 * NU) + k0 + 2 * half);
        acc[t] = WMMA_F32(a, b, acc[t]);
      }
    }
  }
#pragma unroll
  for (int t = 0; t < 4; ++t) {
    const int col = tg * 64 + t * 16 + l16;
    const float bv = b5[col];
#pragma unroll
    for (int i = 0; i < 8; ++i) {
      const int m = tm * 16 + i + 8 * half;
      out[(size_t)m * NU + col] = fast_tanh(acc[t][i] + bv);
    }
  }
}

// ---------------------------------------------------------------------------
extern "C" void kernel_launch(void* const* d_in, const int* in_sizes, int n_in,
                              void* d_out, int out_size, void* d_ws, size_t ws_size,
                              hipStream_t stream) {
  (void)in_sizes; (void)n_in; (void)out_size; (void)ws_size;
  const float* q1  = (const float*)d_in[0];
  const float* q2  = (const float*)d_in[1];
  const float* q3  = (const float*)d_in[2];
  const int*   adj = (const int*)d_in[3];   // jax demotes int64 -> int32 without x64 mode
  const float* WQ  = (const float*)d_in[4];  const float* bQ  = (const float*)d_in[5];
  const float* WV  = (const float*)d_in[6];  const float* bV  = (const float*)d_in[7];
  const float* WQ3 = (const float*)d_in[8];  const float* bQ3 = (const float*)d_in[9];
  const float* WK3 = (const float*)d_in[10]; const float* bK3 = (const float*)d_in[11];
  const float* WV3 = (const float*)d_in[12]; const float* bV3 = (const float*)d_in[13];
  const float* WQ4 = (const float*)d_in[14]; const float* bQ4 = (const float*)d_in[15];
  const float* WV4 = (const float*)d_in[16]; const float* bV4 = (const float*)d_in[17];
  const float* W5  = (const float*)d_in[18]; const float* b5  = (const float*)d_in[19];

  float* out = (float*)d_out;
  float* ws  = (float*)d_ws;
  // Workspace: Q1..Q4 | K3 | V1..V4 (head-split, 9*32MB) | M, 1/Z (4*256KB each) | o4 (32MB)
  float* Qb = ws;
  float* Kb = ws + 4 * PROJ;
  float* Vb = ws + 5 * PROJ;
  float* Mb = ws + 9 * PROJ;
  float* Zb = Mb + 4 * (S * S);
  float* o4 = Zb + 4 * (S * S);

  const dim3 blk(256);
  // 9 projections (gather for Q4/V4 folded in via adj)
  proj_tanh_headsplit<<<1024, blk, 0, stream>>>(q1, nullptr, WQ,  bQ,  Qb + 0 * PROJ);
  proj_tanh_headsplit<<<1024, blk, 0, stream>>>(q2, nullptr, WQ,  bQ,  Qb + 1 * PROJ);
  proj_tanh_headsplit<<<1024, blk, 0, stream>>>(q3, nullptr, WQ3, bQ3, Qb + 2 * PROJ);
  proj_tanh_headsplit<<<1024, blk, 0, stream>>>(q3, adj,     WQ4, bQ4, Qb + 3 * PROJ);
  proj_tanh_headsplit<<<1024, blk, 0, stream>>>(q3, nullptr, WK3, bK3, Kb);
  proj_tanh_headsplit<<<1024, blk, 0, stream>>>(q1, nullptr, WV,  bV,  Vb + 0 * PROJ);
  proj_tanh_headsplit<<<1024, blk, 0, stream>>>(q2, nullptr, WV,  bV,  Vb + 1 * PROJ);
  proj_tanh_headsplit<<<1024, blk, 0, stream>>>(q3, nullptr, WV3, bV3, Vb + 2 * PROJ);
  proj_tanh_headsplit<<<1024, blk, 0, stream>>>(q3, adj,     WV4, bV4, Vb + 3 * PROJ);

  // softmax(axis=0) stats: online max/sum over the 1024 head-batch rows
  attn_stats<<<128, blk, 0, stream>>>(ws, Mb, Zb);
  // attention outputs (o1,o2,o3 -> d_out slots 0,1,3 ; o4 -> workspace)
  attn_out<<<8192, blk, 0, stream>>>(ws, Mb, Zb, out, o4);
  // o5 -> d_out slot 2
  out_proj_tanh<<<1024, blk, 0, stream>>>(out, out + PROJ, out + 3 * PROJ, o4,
                                          W5, b5, out + 2 * PROJ);
}